// GradientGatedDepthContext_51634096832886
// MI455X (gfx1250) — compile-verified
//
#include <hip/hip_runtime.h>
#include <hip/hip_bf16.h>
#include <math.h>

typedef __attribute__((ext_vector_type(2))) float v2f;
typedef __attribute__((ext_vector_type(8))) float v8f;

#define B_  64
#define T_  2048
#define D_  256
#define H_  256
#define EPS 1e-3f

// ---------------------------------------------------------------------------
// Kernel 1: time-parallel projection GEMM  C[M,H] = act(A[M,D] @ W[D,H] (+b))
//   M = B*T = 131072.  Block = 256 threads (8 waves), block tile = 128 x 256.
//   Each wave computes a 16 x 256 strip with V_WMMA_F32_16X16X4_F32,
//   W held in LDS (256 KB, dynamic) with K-pairs interleaved so each B
//   fragment is one aligned 8-byte LDS load.
// ---------------------------------------------------------------------------
template <bool GATE>
__global__ __launch_bounds__(256) void proj_kernel(
    const float* __restrict__ A, const float* __restrict__ W,
    const float* __restrict__ bias, float* __restrict__ C) {
  extern __shared__ float wl[];  // [D/2][2*H]: wl[(k>>1)*2H + n*2 + (k&1)]
  const int tid = threadIdx.x;

  for (int i = tid; i < D_ * H_; i += 256) {
    int k = i / H_, n = i % H_;
    wl[(k >> 1) * (2 * H_) + n * 2 + (k & 1)] = W[i];
  }
  __syncthreads();

  const int wave = tid >> 5;
  const int lane = tid & 31;
  const int l15  = lane & 15;
  const int hi   = lane >> 4;

  const long m0 = (long)blockIdx.x * 128 + wave * 16;  // first row of wave tile

  v8f acc[16];
#pragma unroll
  for (int n = 0; n < 16; ++n) acc[n] = (v8f){0.f, 0.f, 0.f, 0.f, 0.f, 0.f, 0.f, 0.f};

  // A fragment rows: lane l15 -> M=l15; lanes 16-31 hold K+2 of the same rows.
  const float* arow = A + (m0 + l15) * D_;
  for (int k0 = 0; k0 < D_; k0 += 4) {
    v2f a = *(const v2f*)(arow + k0 + 2 * hi);
    const float* wp = &wl[((k0 >> 1) + hi) * (2 * H_)];
#pragma unroll
    for (int n = 0; n < 16; ++n) {
      v2f b = *(const v2f*)(wp + (n * 16 + l15) * 2);
      acc[n] = __builtin_amdgcn_wmma_f32_16x16x4_f32(false, a, false, b,
                                                     (short)0, acc[n], false, false);
    }
  }

  // C/D layout: v8f element e at lane l -> row = e + 8*(l>=16), col = nb + (l&15)
#pragma unroll
  for (int n = 0; n < 16; ++n) {
    const int col = n * 16 + l15;
    float bv = 0.f;
    if constexpr (GATE) bv = bias[col];
#pragma unroll
    for (int e = 0; e < 8; ++e) {
      long row = m0 + e + 8 * hi;
      float v = acc[n][e] + bv;
      if constexpr (GATE) v = 1.f / (1.f + __expf(-v));
      C[row * H_ + col] = v;
    }
  }
}

// ---------------------------------------------------------------------------
// Kernel 2: persistent recurrent scan.  The scan is independent per batch row,
// so 4 blocks x 16 batch rows each, no grid sync.  Per step per block:
//   S = h_prev[16,256] @ Wh[256,256]   (fp32 WMMA, Wh resident in LDS)
//   h  = gate*tanh(xp + S) + (1-gate)*h_prev
//   h  = LayerNorm(h)*gamma + beta     (stats via LDS cross-wave reduction)
// Double-buffered h state in LDS; xp/gate streamed with prefetch.
// ---------------------------------------------------------------------------
__global__ __launch_bounds__(256) void recur_kernel(
    const float* __restrict__ xproj, const float* __restrict__ gate,
    const float* __restrict__ Wh, const float* __restrict__ gamma,
    const float* __restrict__ beta, float* __restrict__ out) {
  extern __shared__ float smem[];
  const int HP = H_ + 2;                    // padded h row stride
  float* wl   = smem;                       // [H/2][2*H] interleaved Wh (256 KB)
  float* hbuf = wl + D_ * H_;               // [2][16][HP] double-buffered state
  float* part = hbuf + 2 * 16 * HP;         // [16][16][2] partial LN sums
  float* mu   = part + 16 * 16 * 2;         // [16]
  float* rs   = mu + 16;                    // [16]

  const int tid = threadIdx.x;
  for (int i = tid; i < H_ * H_; i += 256) {
    int k = i / H_, n = i % H_;
    wl[(k >> 1) * (2 * H_) + n * 2 + (k & 1)] = Wh[i];
  }
  for (int i = tid; i < 2 * 16 * HP; i += 256) hbuf[i] = 0.f;  // h0 = 0
  __syncthreads();

  const int wave = tid >> 5;
  const int lane = tid & 31;
  const int l15  = lane & 15;
  const int hi   = lane >> 4;
  const int b0   = blockIdx.x * 16;   // 16 batch rows per block

  // each wave owns 2 of 16 N-tiles (32 columns)
  const int col0 = (wave * 2) * 16 + l15;
  const int col1 = col0 + 16;
  const float ga0 = gamma[col0], ga1 = gamma[col1];
  const float be0 = beta[col0],  be1 = beta[col1];

  int pb = 0;
  for (int t = 0; t < T_; ++t) {
    const float* hprev = hbuf + pb * 16 * HP;
    float*       hnew  = hbuf + (1 - pb) * 16 * HP;

    // prefetch next timestep's xp/gate tile (16 rows x 256 cols, 64B/thread)
    if (t + 1 < T_) {
      long pbase = ((long)(b0 + (tid & 15)) * T_ + (t + 1)) * H_ + (tid >> 4) * 16;
      __builtin_prefetch(xproj + pbase, 0, 0);
      __builtin_prefetch(gate + pbase, 0, 0);
    }

    v8f acc0 = (v8f){0.f, 0.f, 0.f, 0.f, 0.f, 0.f, 0.f, 0.f};
    v8f acc1 = acc0;
    for (int k0 = 0; k0 < H_; k0 += 4) {
      v2f a = *(const v2f*)(hprev + l15 * HP + k0 + 2 * hi);  // shared A frag
      const float* wp = &wl[((k0 >> 1) + hi) * (2 * H_)];
      v2f bv0 = *(const v2f*)(wp + col0 * 2);
      v2f bv1 = *(const v2f*)(wp + col1 * 2);
      acc0 = __builtin_amdgcn_wmma_f32_16x16x4_f32(false, a, false, bv0,
                                                   (short)0, acc0, false, false);
      acc1 = __builtin_amdgcn_wmma_f32_16x16x4_f32(false, a, false, bv1,
                                                   (short)0, acc1, false, false);
    }

    // gate + tanh, write raw h to the new buffer for LN stats
    float hv0[8], hv1[8];
#pragma unroll
    for (int e = 0; e < 8; ++e) {
      const int m = e + 8 * hi;
      const long base = ((long)(b0 + m) * T_ + t) * H_;
      float xp0 = xproj[base + col0], g0 = gate[base + col0];
      float xp1 = xproj[base + col1], g1 = gate[base + col1];
      float hp0 = hprev[m * HP + col0], hp1 = hprev[m * HP + col1];
      float c0 = tanhf(xp0 + acc0[e]);
      float c1 = tanhf(xp1 + acc1[e]);
      float h0 = g0 * c0 + (1.f - g0) * hp0;
      float h1 = g1 * c1 + (1.f - g1) * hp1;
      hv0[e] = h0; hv1[e] = h1;
      hnew[m * HP + col0] = h0;
      hnew[m * HP + col1] = h1;
    }
    __syncthreads();

    // LayerNorm stats: 256 threads -> row = tid>>4, 16 partial cols each
    {
      const int row = tid >> 4, sub = tid & 15;
      float s = 0.f, ss = 0.f;
      const float* hr = hnew + row * HP;
#pragma unroll
      for (int c = 0; c < 16; ++c) {
        float v = hr[sub + c * 16];
        s += v; ss += v * v;
      }
      part[(row * 16 + sub) * 2]     = s;
      part[(row * 16 + sub) * 2 + 1] = ss;
    }
    __syncthreads();
    if (tid < 16) {
      float s = 0.f, ss = 0.f;
#pragma unroll
      for (int i = 0; i < 16; ++i) {
        s  += part[(tid * 16 + i) * 2];
        ss += part[(tid * 16 + i) * 2 + 1];
      }
      float m_ = s * (1.f / H_);
      float v_ = ss * (1.f / H_) - m_ * m_;
      mu[tid] = m_;
      rs[tid] = rsqrtf(v_ + EPS);
    }
    __syncthreads();

    // normalize from registers, emit output, publish normalized state
#pragma unroll
    for (int e = 0; e < 8; ++e) {
      const int m = e + 8 * hi;
      const float mm = mu[m], rr = rs[m];
      float o0 = (hv0[e] - mm) * rr * ga0 + be0;
      float o1 = (hv1[e] - mm) * rr * ga1 + be1;
      const long base = ((long)(b0 + m) * T_ + t) * H_;
      out[base + col0] = o0;
      out[base + col1] = o1;
      hnew[m * HP + col0] = o0;
      hnew[m * HP + col1] = o1;
    }
    __syncthreads();
    pb ^= 1;
  }
}

// ---------------------------------------------------------------------------
extern "C" void kernel_launch(void* const* d_in, const int* in_sizes, int n_in,
                              void* d_out, int out_size, void* d_ws, size_t ws_size,
                              hipStream_t stream) {
  const float* x_seq = (const float*)d_in[0];
  const float* g_seq = (const float*)d_in[1];
  const float* Wg    = (const float*)d_in[2];
  const float* bg    = (const float*)d_in[3];
  const float* Wx    = (const float*)d_in[4];
  const float* Wh    = (const float*)d_in[5];
  const float* gamma = (const float*)d_in[6];
  const float* beta  = (const float*)d_in[7];
  float* out = (float*)d_out;

  float* xproj = (float*)d_ws;                       // [B*T, H]
  float* gatep = xproj + (size_t)B_ * T_ * H_;       // [B*T, H]

  const size_t lds1 = (size_t)D_ * H_ * sizeof(float);            // 256 KB
  const int blocks1 = (B_ * T_) / 128;

  proj_kernel<false><<<blocks1, 256, lds1, stream>>>(x_seq, Wx, nullptr, xproj);
  proj_kernel<true><<<blocks1, 256, lds1, stream>>>(g_seq, Wg, bg, gatep);

  const size_t lds2 =
      ((size_t)D_ * H_ + 2 * 16 * (H_ + 2) + 16 * 16 * 2 + 32) * sizeof(float);
  recur_kernel<<<B_ / 16, 256, lds2, stream>>>(xproj, gatep, Wh, gamma, beta, out);
}